// get_model_32504312496500
// MI455X (gfx1250) — compile-verified
//
// MI455X / gfx1250 implementation of the PointNet++-style autoencoder forward.
// All linear algebra goes through V_WMMA_F32_16X16X4_F32 (fp32 matrix pipe,
// matching the fp32 reference). GEMMs are template-specialized on
// (transB, activation, tile-count) so hot loops have no runtime flag branches,
// and the wide path computes a 16x64 strip per wave (4 back-to-back WMMAs per
// k-step, 4x A-fragment reuse). Everything launches on `stream`; scratch is a
// bump allocator over d_ws (~155 MB used).
//
// Input pointer order assumes setup_inputs() dict insertion order, with the
// nested `params` tree flattened in insertion order:
//   0 xyz, 1 noise,
//   sa1[3]{W,b,g,be}, sa2[3]{W,b,g,be},
//   fe1{Wpos,bpos,Wq,bq,Wk,bk,Wv,bv,Wt,bt}, fe2{...},
//   dec{W1,b1,W2,b2}, coor{W1,b1,W2,b2,W3,b3},
//   up[4]{Wo1,bo1,Wo2,bo2,Wf,bf}            -> n_in == 80

#include <hip/hip_runtime.h>

typedef __attribute__((ext_vector_type(2))) float v2f;
typedef __attribute__((ext_vector_type(8))) float v8f;

#define ACT_NONE 0
#define ACT_RELU 1
#define ACT_TANH 2

__device__ __forceinline__ float apply_act(float v, int ACTV) {
  if (ACTV == ACT_RELU) return fmaxf(v, 0.f);
  if (ACTV == ACT_TANH) return tanhf(v);
  return v;
}

// ---------------------------------------------------------------------------
// Generic fp32 WMMA GEMM: C[M,N] = act( (A - Asub)[M,K] * B + bias ) + resid
//   TRANSB=1: B element (k,n) = Bm[n*ldb + k]  (weight layout W[out,in])
//   TRANSB=0: B element (k,n) = Bm[k*ldb + n]  (plain row-major, e.g. V, attn)
// NT = 16-wide N-tiles per wave (1 or 4). NT==4 requires K%4==0 && N%64==0;
// the M edge is handled with clamped A rows + epilogue bounds checks.
// A fragment layout (16x4 f32): lanes 0-15 hold K={k,k+1}, lanes 16-31 K={k+2,k+3}.
// C/D layout: VGPR j, lanes 0-15 -> (m0+j, n), lanes 16-31 -> (m0+8+j, n-16).
// ---------------------------------------------------------------------------
template <bool TRANSB, int ACTV, int NT>
__global__ void __launch_bounds__(32)
gemm_wmma_kernel(const float* __restrict__ A, const float* __restrict__ Asub,
                 const float* __restrict__ Bm, const float* __restrict__ bias,
                 const float* __restrict__ resid, float* __restrict__ C,
                 int M, int N, int K, int lda, int ldb, int ldc,
                 long bA, long bB, long bC, long bR) {
  const int z = blockIdx.z;
  const float* Ab = A + (long)z * bA;
  const float* Sb = Asub ? Asub + (long)z * bA : nullptr;
  const float* Bb = Bm + (long)z * bB;
  const float* Rb = resid ? resid + (long)z * bR : nullptr;
  float* Cb = C + (long)z * bC;

  const int lane = threadIdx.x & 31;
  const int half = lane >> 4;  // 0: lanes 0-15, 1: lanes 16-31
  const int lm = lane & 15;
  const int m0 = blockIdx.y * 16;
  const int n0 = blockIdx.x * (16 * NT);

  v8f acc[NT];
#pragma unroll
  for (int t = 0; t < NT; ++t) acc[t] = v8f{0.f, 0.f, 0.f, 0.f, 0.f, 0.f, 0.f, 0.f};

  if (NT == 4) {
    // Dispatch guarantees K%4==0 and N%64==0; clamp A rows for the M edge.
    int rowc = m0 + lm;
    if (rowc >= M) rowc = M - 1;
    const float* ap = Ab + (long)rowc * lda + 2 * half;
    const float* sp = Sb ? Sb + (long)rowc * lda + 2 * half : nullptr;
    if (TRANSB) {
      const float* bp0 = Bb + (long)(n0 + lm) * ldb + 2 * half;
      const long tstep = (long)16 * ldb;
      for (int k = 0; k < K; k += 4) {
        if ((k & 31) == 0) __builtin_prefetch(ap + k + 128, 0, 2);
        v2f a;
        a.x = ap[k]; a.y = ap[k + 1];
        if (sp) { a.x -= sp[k]; a.y -= sp[k + 1]; }
#pragma unroll
        for (int t = 0; t < 4; ++t) {
          v2f b;
          b.x = bp0[(long)t * tstep + k];
          b.y = bp0[(long)t * tstep + k + 1];
          acc[t] = __builtin_amdgcn_wmma_f32_16x16x4_f32(false, a, false, b,
                                                         (short)0, acc[t], false, false);
        }
      }
    } else {
      const float* bp0 = Bb + (long)(2 * half) * ldb + n0 + lm;
      for (int k = 0; k < K; k += 4) {
        v2f a;
        a.x = ap[k]; a.y = ap[k + 1];
        if (sp) { a.x -= sp[k]; a.y -= sp[k + 1]; }
        const float* bk = bp0 + (long)k * ldb;
#pragma unroll
        for (int t = 0; t < 4; ++t) {
          v2f b;
          b.x = bk[16 * t];
          b.y = bk[(long)ldb + 16 * t];
          acc[t] = __builtin_amdgcn_wmma_f32_16x16x4_f32(false, a, false, b,
                                                         (short)0, acc[t], false, false);
        }
      }
    }
  } else {
    const int row = m0 + lm, col = n0 + lm;
    const int Kmain = K & ~3;
    if (m0 + 16 <= M && n0 + 16 <= N) {
      const float* ap = Ab + (long)row * lda + 2 * half;
      const float* sp = Sb ? Sb + (long)row * lda + 2 * half : nullptr;
      for (int k = 0; k < Kmain; k += 4) {
        v2f a, b;
        a.x = ap[k]; a.y = ap[k + 1];
        if (sp) { a.x -= sp[k]; a.y -= sp[k + 1]; }
        const int ka = k + 2 * half;
        if (TRANSB) { b.x = Bb[(long)col * ldb + ka];  b.y = Bb[(long)col * ldb + ka + 1]; }
        else        { b.x = Bb[(long)ka * ldb + col];  b.y = Bb[(long)(ka + 1) * ldb + col]; }
        acc[0] = __builtin_amdgcn_wmma_f32_16x16x4_f32(false, a, false, b,
                                                       (short)0, acc[0], false, false);
      }
      if (K & 3) {  // single guarded tail step
        const int ka = Kmain + 2 * half;
        v2f a = {0.f, 0.f}, b = {0.f, 0.f};
        if (ka < K)     { a.x = ap[Kmain];     if (sp) a.x -= sp[Kmain]; }
        if (ka + 1 < K) { a.y = ap[Kmain + 1]; if (sp) a.y -= sp[Kmain + 1]; }
        if (TRANSB) {
          if (ka < K)     b.x = Bb[(long)col * ldb + ka];
          if (ka + 1 < K) b.y = Bb[(long)col * ldb + ka + 1];
        } else {
          if (ka < K)     b.x = Bb[(long)ka * ldb + col];
          if (ka + 1 < K) b.y = Bb[(long)(ka + 1) * ldb + col];
        }
        acc[0] = __builtin_amdgcn_wmma_f32_16x16x4_f32(false, a, false, b,
                                                       (short)0, acc[0], false, false);
      }
    } else {
      for (int k = 0; k < K; k += 4) {
        const int ka = k + 2 * half;
        v2f a = {0.f, 0.f}, b = {0.f, 0.f};
        if (row < M) {
          if (ka < K) {
            a.x = Ab[(long)row * lda + ka];
            if (Sb) a.x -= Sb[(long)row * lda + ka];
          }
          if (ka + 1 < K) {
            a.y = Ab[(long)row * lda + ka + 1];
            if (Sb) a.y -= Sb[(long)row * lda + ka + 1];
          }
        }
        if (col < N) {
          if (TRANSB) {
            if (ka < K)     b.x = Bb[(long)col * ldb + ka];
            if (ka + 1 < K) b.y = Bb[(long)col * ldb + ka + 1];
          } else {
            if (ka < K)     b.x = Bb[(long)ka * ldb + col];
            if (ka + 1 < K) b.y = Bb[(long)(ka + 1) * ldb + col];
          }
        }
        acc[0] = __builtin_amdgcn_wmma_f32_16x16x4_f32(false, a, false, b,
                                                       (short)0, acc[0], false, false);
      }
    }
  }

#pragma unroll
  for (int t = 0; t < NT; ++t) {
    const int c = n0 + 16 * t + lm;
#pragma unroll
    for (int j = 0; j < 8; ++j) {
      const int r = m0 + j + 8 * half;
      if (r < M && c < N) {
        float v = acc[t][j];
        if (bias) v += bias[c];
        v = apply_act(v, ACTV);
        if (Rb) v += Rb[(long)r * ldc + c];
        Cb[(long)r * ldc + c] = v;
      }
    }
  }
}

// ---------------------------------------------------------------------------
// Farthest point sampling: one block per batch; serial over npoint picks,
// parallel distance update + LDS argmax tree-reduce each pick.
// ---------------------------------------------------------------------------
#define FPS_T 512
__global__ void fps_kernel(const float* __restrict__ xyz, int* __restrict__ fidx,
                           int N, int npoint) {
  const int b = blockIdx.x, tid = threadIdx.x;
  __shared__ float s_val[FPS_T];
  __shared__ int s_idx[FPS_T];
  __shared__ int s_far;
  const float* P = xyz + (long)b * N * 3;
  const int ppt = (N + FPS_T - 1) / FPS_T;  // <= 16
  float dist[16];
  for (int j = 0; j < ppt; ++j) dist[j] = 1e10f;
  if (tid == 0) s_far = 0;
  __syncthreads();
  for (int i = 0; i < npoint; ++i) {
    const int far = s_far;
    if (tid == 0) fidx[b * npoint + i] = far;
    const float cx = P[far * 3], cy = P[far * 3 + 1], cz = P[far * 3 + 2];
    float bv = -1.f; int bi = 0;
    for (int j = 0; j < ppt; ++j) {
      const int p = tid + j * FPS_T;
      if (p < N) {
        const float dx = P[p * 3] - cx, dy = P[p * 3 + 1] - cy, dz = P[p * 3 + 2] - cz;
        const float d = dx * dx + dy * dy + dz * dz;
        if (d < dist[j]) dist[j] = d;
        if (dist[j] > bv) { bv = dist[j]; bi = p; }
      }
    }
    s_val[tid] = bv; s_idx[tid] = bi;
    __syncthreads();
    for (int o = FPS_T / 2; o > 0; o >>= 1) {
      if (tid < o) {
        if (s_val[tid + o] > s_val[tid] ||
            (s_val[tid + o] == s_val[tid] && s_idx[tid + o] < s_idx[tid])) {
          s_val[tid] = s_val[tid + o]; s_idx[tid] = s_idx[tid + o];
        }
      }
      __syncthreads();
    }
    if (tid == 0) s_far = s_idx[0];
    __syncthreads();
  }
}

__global__ void gather_xyz_kernel(const float* __restrict__ xyz, const int* __restrict__ fidx,
                                  float* __restrict__ out, int N, int S, int BS) {
  const int q = blockIdx.x * blockDim.x + threadIdx.x;
  if (q >= BS) return;
  const int b = q / S;
  int id = fidx[q];
  if (id < 0) id = 0;
  if (id >= N) id = N - 1;
  const float* src = xyz + ((long)b * N + id) * 3;
  out[q * 3] = src[0]; out[q * 3 + 1] = src[1]; out[q * 3 + 2] = src[2];
}

// Ball query: one wave per query point; ballot/popc stream-compaction keeps the
// first K passing indices in increasing order, pads with the first hit.
__global__ void ball_query_kernel(const float* __restrict__ xyz, const float* __restrict__ nxyz,
                                  int* __restrict__ idx, int BS, int S, int N, int K, float r2) {
  const int q = blockIdx.x * (blockDim.x >> 5) + (threadIdx.x >> 5);
  const int lane = threadIdx.x & 31;
  if (q >= BS) return;
  const int b = q / S;
  const float* P = xyz + (long)b * N * 3;
  const float cx = nxyz[q * 3], cy = nxyz[q * 3 + 1], cz = nxyz[q * 3 + 2];
  int found = 0, first = -1;
  for (int base = 0; base < N; base += 32) {
    if (found >= K) break;
    const int p = base + lane;
    bool pass = false;
    if (p < N) {
      const float dx = P[p * 3] - cx, dy = P[p * 3 + 1] - cy, dz = P[p * 3 + 2] - cz;
      pass = (dx * dx + dy * dy + dz * dz) <= r2;
    }
    const unsigned long long m = __ballot(pass);
    const int rank = __popcll(m & ((1ull << lane) - 1ull));
    if (pass && (found + rank) < K) idx[(long)q * K + found + rank] = p;
    if (first < 0 && m) first = base + (__ffsll((unsigned long long)m) - 1);
    found += __popcll(m);
  }
  if (first < 0) first = 0;
  for (int t = found + lane; t < K; t += 32) idx[(long)q * K + t] = first;
}

// feat[b,s,k,:] = concat( xyz[idx]-new_xyz , pts[idx] )
__global__ void gather_feat_kernel(const float* __restrict__ xyz, const float* __restrict__ nxyz,
                                   const float* __restrict__ pts, int Cpts,
                                   const int* __restrict__ idx, float* __restrict__ feat,
                                   int BS, int S, int K, int N, int Cout) {
  const long t = (long)blockIdx.x * blockDim.x + threadIdx.x;
  if (t >= (long)BS * K) return;
  const int q = (int)(t / K);
  const int b = q / S;
  int id = idx[t];
  if (id < 0) id = 0;
  if (id >= N) id = N - 1;
  const float* psrc = xyz + ((long)b * N + id) * 3;
  float* dst = feat + t * Cout;
  dst[0] = psrc[0] - nxyz[q * 3];
  dst[1] = psrc[1] - nxyz[q * 3 + 1];
  dst[2] = psrc[2] - nxyz[q * 3 + 2];
  const float* fsrc = pts + ((long)b * N + id) * Cpts;
  for (int c = 0; c < Cpts; ++c) dst[3 + c] = fsrc[c];
}

// Batch-norm statistics: one block per channel.
__global__ void bn_stats_kernel(const float* __restrict__ X, long M, int N,
                                float* __restrict__ mean, float* __restrict__ var) {
  const int n = blockIdx.x, tid = threadIdx.x;
  __shared__ float ss[256], sq[256];
  float s = 0.f, q = 0.f;
  for (long r = tid; r < M; r += blockDim.x) {
    const float v = X[r * N + n];
    s += v; q += v * v;
  }
  ss[tid] = s; sq[tid] = q;
  __syncthreads();
  for (int o = blockDim.x / 2; o > 0; o >>= 1) {
    if (tid < o) { ss[tid] += ss[tid + o]; sq[tid] += sq[tid + o]; }
    __syncthreads();
  }
  if (tid == 0) {
    const float m = ss[0] / (float)M;
    mean[n] = m;
    var[n] = sq[0] / (float)M - m * m;
  }
}

__global__ void bn_relu_kernel(float* __restrict__ X, const float* __restrict__ mean,
                               const float* __restrict__ var, const float* __restrict__ g,
                               const float* __restrict__ be, long total, int N) {
  const long i = (long)blockIdx.x * blockDim.x + threadIdx.x;
  if (i >= total) return;
  const int n = (int)(i % N);
  const float v = g[n] * (X[i] - mean[n]) * rsqrtf(var[n] + 1e-5f) + be[n];
  X[i] = fmaxf(v, 0.f);
}

// Y[g,n] = max_k X[g,k,n]
__global__ void maxpool_kernel(const float* __restrict__ X, float* __restrict__ Y,
                               int G, int K, int N) {
  const long t = (long)blockIdx.x * blockDim.x + threadIdx.x;
  if (t >= (long)G * N) return;
  const int g = (int)(t / N), n = (int)(t % N);
  float m = -3.4e38f;
  for (int k = 0; k < K; ++k) m = fmaxf(m, X[((long)g * K + k) * N + n]);
  Y[t] = m;
}

// Row softmax, one wave per row.
__global__ void softmax_rows_kernel(float* __restrict__ X, int rows, int C) {
  const int row = blockIdx.x * (blockDim.x >> 5) + (threadIdx.x >> 5);
  const int lane = threadIdx.x & 31;
  if (row >= rows) return;
  float* x = X + (long)row * C;
  float mx = -3.4e38f;
  for (int c = lane; c < C; c += 32) mx = fmaxf(mx, x[c]);
  for (int o = 16; o; o >>= 1) mx = fmaxf(mx, __shfl_xor(mx, o, 32));
  float s = 0.f;
  for (int c = lane; c < C; c += 32) { const float e = expf(x[c] - mx); x[c] = e; s += e; }
  for (int o = 16; o; o >>= 1) s += __shfl_xor(s, o, 32);
  const float inv = 1.f / s;
  for (int c = lane; c < C; c += 32) x[c] *= inv;
}

__global__ void add_noise_kernel(const float* __restrict__ x, const float* __restrict__ noise,
                                 float* __restrict__ comp, float* __restrict__ outcopy, int n) {
  const int i = blockIdx.x * blockDim.x + threadIdx.x;
  if (i >= n) return;
  const float v = x[i] + noise[i];
  comp[i] = v;
  outcopy[i] = v;
}

// out[b, 2m+r] = xyz[b,m] + radius * off[b,m,r]; flat: p = b*M+m -> 2p+r
__global__ void upsample_coords_kernel(const float* __restrict__ xyz, const float* __restrict__ off,
                                       float* __restrict__ out, int P, float radius) {
  const int t = blockIdx.x * blockDim.x + threadIdx.x;
  if (t >= 2 * P) return;
  const int p = t >> 1, r = t & 1;
  out[t * 3]     = xyz[p * 3]     + radius * off[p * 6 + r * 3];
  out[t * 3 + 1] = xyz[p * 3 + 1] + radius * off[p * 6 + r * 3 + 1];
  out[t * 3 + 2] = xyz[p * 3 + 2] + radius * off[p * 6 + r * 3 + 2];
}

// Chamfer: sum over rows of min_j ||a_r - b_j||^2. B points staged through LDS
// in 2048-point chunks; wave-per-row min + shfl reduce; block atomicAdd.
#define CH_PTS 2048
__global__ void chamfer_min_kernel(const float* __restrict__ Ap, const float* __restrict__ Bp,
                                   int Na, int Nb, float* __restrict__ sum_out) {
  __shared__ float sB[CH_PTS * 3];
  __shared__ float blkSum;
  const int b = blockIdx.y;
  const float* Abase = Ap + (long)b * Na * 3;
  const float* Bbase = Bp + (long)b * Nb * 3;
  if (threadIdx.x == 0) blkSum = 0.f;
  const int wave = threadIdx.x >> 5, lane = threadIdx.x & 31;
  const int row0 = blockIdx.x * 64 + wave * 8;
  float rmin[8];
#pragma unroll
  for (int i = 0; i < 8; ++i) rmin[i] = 3.4e38f;
  for (int c0 = 0; c0 < Nb; c0 += CH_PTS) {
    const int cnt = (Nb - c0 < CH_PTS) ? (Nb - c0) : CH_PTS;
    __syncthreads();
    for (int i = threadIdx.x; i < cnt * 3; i += blockDim.x) sB[i] = Bbase[c0 * 3 + i];
    __syncthreads();
    for (int rr = 0; rr < 8; ++rr) {
      const int r = row0 + rr;
      if (r >= Na) break;
      const float ax = Abase[r * 3], ay = Abase[r * 3 + 1], az = Abase[r * 3 + 2];
      float mn = rmin[rr];
      for (int j = lane; j < cnt; j += 32) {
        const float dx = ax - sB[j * 3], dy = ay - sB[j * 3 + 1], dz = az - sB[j * 3 + 2];
        mn = fminf(mn, dx * dx + dy * dy + dz * dz);
      }
      rmin[rr] = mn;
    }
  }
  float acc = 0.f;
  for (int rr = 0; rr < 8; ++rr) {
    const int r = row0 + rr;
    if (r >= Na) break;
    float mn = rmin[rr];
    for (int o = 16; o; o >>= 1) mn = fminf(mn, __shfl_xor(mn, o, 32));
    if (lane == 0) acc += mn;
  }
  if (lane == 0) atomicAdd(&blkSum, acc);
  __syncthreads();
  if (threadIdx.x == 0) atomicAdd(sum_out, blkSum);
}

__global__ void zero_kernel(float* __restrict__ p, int n) {
  const int i = blockIdx.x * blockDim.x + threadIdx.x;
  if (i < n) p[i] = 0.f;
}

__global__ void chamfer_final_kernel(const float* __restrict__ sums, float* __restrict__ out,
                                     float inv1, float inv2) {
  if (threadIdx.x == 0 && blockIdx.x == 0) out[0] = sums[0] * inv1 + sums[1] * inv2;
}

// ---------------------------------------------------------------------------
// Host-side orchestration
// ---------------------------------------------------------------------------
static inline dim3 g1(long n, int t) { return dim3((unsigned)((n + t - 1) / t)); }

template <bool TB, int ACTV>
static void launch_gemm_t(hipStream_t st, const float* A, const float* Asub, const float* Bm,
                          const float* bias, const float* resid, float* C,
                          int M, int N, int K, int lda, int ldb, int ldc,
                          int nb, long bA, long bB, long bC, long bR) {
  if ((K & 3) == 0 && (N & 63) == 0) {
    dim3 grid(N / 64, (M + 15) / 16, nb);
    gemm_wmma_kernel<TB, ACTV, 4><<<grid, 32, 0, st>>>(A, Asub, Bm, bias, resid, C,
                                                       M, N, K, lda, ldb, ldc, bA, bB, bC, bR);
  } else {
    dim3 grid((N + 15) / 16, (M + 15) / 16, nb);
    gemm_wmma_kernel<TB, ACTV, 1><<<grid, 32, 0, st>>>(A, Asub, Bm, bias, resid, C,
                                                       M, N, K, lda, ldb, ldc, bA, bB, bC, bR);
  }
}

static void run_gemm(hipStream_t st, const float* A, const float* Asub, const float* Bm,
                     const float* bias, const float* resid, float* C,
                     int M, int N, int K, int lda, int ldb, int ldc,
                     int nb, long bA, long bB, long bC, long bR, int transB, int act) {
  if (transB) {
    if (act == ACT_NONE)      launch_gemm_t<true, ACT_NONE>(st, A, Asub, Bm, bias, resid, C, M, N, K, lda, ldb, ldc, nb, bA, bB, bC, bR);
    else if (act == ACT_RELU) launch_gemm_t<true, ACT_RELU>(st, A, Asub, Bm, bias, resid, C, M, N, K, lda, ldb, ldc, nb, bA, bB, bC, bR);
    else                      launch_gemm_t<true, ACT_TANH>(st, A, Asub, Bm, bias, resid, C, M, N, K, lda, ldb, ldc, nb, bA, bB, bC, bR);
  } else {
    if (act == ACT_NONE)      launch_gemm_t<false, ACT_NONE>(st, A, Asub, Bm, bias, resid, C, M, N, K, lda, ldb, ldc, nb, bA, bB, bC, bR);
    else if (act == ACT_RELU) launch_gemm_t<false, ACT_RELU>(st, A, Asub, Bm, bias, resid, C, M, N, K, lda, ldb, ldc, nb, bA, bB, bC, bR);
    else                      launch_gemm_t<false, ACT_TANH>(st, A, Asub, Bm, bias, resid, C, M, N, K, lda, ldb, ldc, nb, bA, bB, bC, bR);
  }
}

struct FEParams { const float *Wpos, *bpos, *Wq, *bq, *Wk, *bk, *Wv, *bv, *Wt, *bt; };

static void run_fe(hipStream_t st, const float* xyzp, const float* feat, const FEParams& P,
                   float* out, float* x, float* q, float* k, float* v, float* sc, float* xr,
                   int Bb, int S, int C, int C4) {
  const int BS = Bb * S;
  // x = feat + lin(xyz, Wpos)
  run_gemm(st, xyzp, nullptr, P.Wpos, P.bpos, feat, x, BS, C, 3, 3, 3, C,
           1, 0, 0, 0, 0, 1, ACT_NONE);
  run_gemm(st, x, nullptr, P.Wq, P.bq, nullptr, q, BS, C4, C, C, C, C4, 1, 0, 0, 0, 0, 1, ACT_NONE);
  run_gemm(st, x, nullptr, P.Wk, P.bk, nullptr, k, BS, C4, C, C, C, C4, 1, 0, 0, 0, 0, 1, ACT_NONE);
  run_gemm(st, x, nullptr, P.Wv, P.bv, nullptr, v, BS, C,  C, C, C, C,  1, 0, 0, 0, 0, 1, ACT_NONE);
  // scores[b] = q[b] @ k[b]^T   (k plays the "weight" role -> transB=1)
  run_gemm(st, q, nullptr, k, nullptr, nullptr, sc, S, S, C4, C4, C4, S,
           Bb, (long)S * C4, (long)S * C4, (long)S * S, 0, 1, ACT_NONE);
  softmax_rows_kernel<<<g1(BS, 4), 128, 0, st>>>(sc, BS, S);
  // x_r[b] = attn[b] @ v[b]    (plain row-major -> transB=0)
  run_gemm(st, sc, nullptr, v, nullptr, nullptr, xr, S, C, S, S, C, C,
           Bb, (long)S * S, (long)S * C, (long)S * C, 0, 0, ACT_NONE);
  // out = x + relu(lin(x - xr, Wt))
  run_gemm(st, x, xr, P.Wt, P.bt, x, out, BS, C, C, C, C, C, 1, 0, 0, 0, 0, 1, ACT_RELU);
}

static void run_sa(hipStream_t st, const float* in_xyz, int Npts, const float* in_pts, int Cpts,
                   int S, int Ksamp, float radius,
                   const float* const W[3], const float* const bb[3],
                   const float* const gg[3], const float* const be[3], const int Couts[3],
                   int* fidx, float* new_xyz, int* bidx, float* bufA, float* bufB,
                   float* mean, float* var, float* out_f) {
  const int Bb = 8;
  const int BS = Bb * S;
  fps_kernel<<<Bb, FPS_T, 0, st>>>(in_xyz, fidx, Npts, S);
  gather_xyz_kernel<<<g1(BS, 256), 256, 0, st>>>(in_xyz, fidx, new_xyz, Npts, S, BS);
  ball_query_kernel<<<g1(BS, 4), 128, 0, st>>>(in_xyz, new_xyz, bidx, BS, S, Npts,
                                               Ksamp, radius * radius);
  const long Mrows = (long)BS * Ksamp;
  int Cin = 3 + Cpts;
  gather_feat_kernel<<<g1(Mrows, 256), 256, 0, st>>>(in_xyz, new_xyz, in_pts, Cpts,
                                                     bidx, bufA, BS, S, Ksamp, Npts, Cin);
  float* cur = bufA;
  for (int l = 0; l < 3; ++l) {
    const int Cout = Couts[l];
    float* o = (cur == bufA) ? bufB : bufA;
    run_gemm(st, cur, nullptr, W[l], bb[l], nullptr, o, (int)Mrows, Cout, Cin,
             Cin, Cin, Cout, 1, 0, 0, 0, 0, 1, ACT_NONE);
    bn_stats_kernel<<<Cout, 256, 0, st>>>(o, Mrows, Cout, mean, var);
    bn_relu_kernel<<<g1(Mrows * Cout, 256), 256, 0, st>>>(o, mean, var, gg[l], be[l],
                                                          Mrows * Cout, Cout);
    cur = o; Cin = Cout;
  }
  maxpool_kernel<<<g1((long)BS * Couts[2], 256), 256, 0, st>>>(cur, out_f, BS, Ksamp, Couts[2]);
}

extern "C" void kernel_launch(void* const* d_in, const int* in_sizes, int n_in,
                              void* d_out, int out_size, void* d_ws, size_t ws_size,
                              hipStream_t stream) {
  (void)in_sizes; (void)n_in; (void)out_size; (void)ws_size;
  const int B = 8, NPTS = 8192, S1 = 512, K1 = 32, S2 = 128, K2 = 64, NP0 = 128;

  auto F = [&](int i) { return (const float*)d_in[i]; };
  const float* xyz = F(0);    // (8,8192,3)
  const float* noise = F(1);  // (8,256)
  int p = 2;
  const float *sa1W[3], *sa1b[3], *sa1g[3], *sa1be[3];
  for (int l = 0; l < 3; ++l) { sa1W[l] = F(p++); sa1b[l] = F(p++); sa1g[l] = F(p++); sa1be[l] = F(p++); }
  const float *sa2W[3], *sa2b[3], *sa2g[3], *sa2be[3];
  for (int l = 0; l < 3; ++l) { sa2W[l] = F(p++); sa2b[l] = F(p++); sa2g[l] = F(p++); sa2be[l] = F(p++); }
  FEParams fe1{F(p), F(p+1), F(p+2), F(p+3), F(p+4), F(p+5), F(p+6), F(p+7), F(p+8), F(p+9)}; p += 10;
  FEParams fe2{F(p), F(p+1), F(p+2), F(p+3), F(p+4), F(p+5), F(p+6), F(p+7), F(p+8), F(p+9)}; p += 10;
  const float *decW1 = F(p++), *decb1 = F(p++), *decW2 = F(p++), *decb2 = F(p++);
  const float *coW1 = F(p++), *cob1 = F(p++), *coW2 = F(p++), *cob2 = F(p++),
              *coW3 = F(p++), *cob3 = F(p++);
  struct UPParams { const float *Wo1, *bo1, *Wo2, *bo2, *Wf, *bf; } up[4];
  for (int u = 0; u < 4; ++u) { up[u] = {F(p), F(p+1), F(p+2), F(p+3), F(p+4), F(p+5)}; p += 6; }

  // ---- workspace bump allocator (~155 MB of d_ws) ----
  float* wsf = (float*)d_ws;
  size_t off = 0;
  auto allocf = [&](size_t n) { float* r = wsf + off; off += (n + 63) & ~(size_t)63; return r; };
  auto alloci = [&](size_t n) { return (int*)allocf(n); };

  int* fidx1 = alloci(B * S1);
  float* l1_xyz = allocf(B * S1 * 3);
  int* bidx1 = alloci((size_t)B * S1 * K1);
  int* fidx2 = alloci(B * S2);
  float* l2_xyz = allocf(B * S2 * 3);
  int* bidx2 = alloci((size_t)B * S2 * K2);
  float* bufA = allocf((size_t)B * S2 * K2 * 131);   // 8.59M f (covers 131072x64 too)
  float* bufB = allocf((size_t)B * S1 * K1 * 128);   // 16.78M f (covers 65536x256 too)
  float* mean = allocf(256);
  float* var = allocf(256);
  float* l1_f = allocf((size_t)B * S1 * 128);
  float* l2_f = allocf((size_t)B * S2 * 256);
  // feature-enhance scratch (shared by fe1/fe2): C=256, C4=64, S=128
  float* fe_x = allocf((size_t)B * S2 * 256);
  float* fe_q = allocf((size_t)B * S2 * 64);
  float* fe_k = allocf((size_t)B * S2 * 64);
  float* fe_v = allocf((size_t)B * S2 * 256);
  float* fe_sc = allocf((size_t)B * S2 * S2);
  float* fe_xr = allocf((size_t)B * S2 * 256);
  float* xfe1 = allocf((size_t)B * S2 * 256);
  float* xmax = allocf(B * 256);
  float* comp = allocf(B * 256);
  float* H = allocf((size_t)B * NP0 * 256);
  float* dec_feat = allocf((size_t)B * NP0 * 256);
  float* ct1 = allocf((size_t)B * NP0 * 128);
  float* ct2 = allocf((size_t)B * NP0 * 128);
  float* nxyz0 = allocf((size_t)B * NP0 * 3);
  float* featU0 = allocf((size_t)B * NP0 * 256);
  float* hbuf = allocf((size_t)B * 1024 * 128);
  float* offbuf = allocf((size_t)B * 1024 * 6);
  float* childA = allocf((size_t)B * 1024 * 512);
  float* childB = allocf((size_t)B * 1024 * 512);
  float* upxyzA = allocf((size_t)B * 1024 * 3);
  float* upxyzB = allocf((size_t)B * 1024 * 3);
  float* sums = allocf(64);

  float* out_xyz = (float*)d_out;              // (8,2048,3) = 49152
  float* out_cd = out_xyz + 49152;             // scalar
  float* out_comp = out_xyz + 49153;           // (8,256)

  // ---- set abstraction 1: 8192 -> 512, K=32, 6->64->64->128 ----
  { const int Couts[3] = {64, 64, 128};
    run_sa(stream, xyz, NPTS, xyz, 3, S1, K1, 0.2f, sa1W, sa1b, sa1g, sa1be, Couts,
           fidx1, l1_xyz, bidx1, bufA, bufB, mean, var, l1_f); }
  // ---- set abstraction 2: 512 -> 128, K=64, 131->128->128->256 ----
  { const int Couts[3] = {128, 128, 256};
    run_sa(stream, l1_xyz, S1, l1_f, 128, S2, K2, 0.4f, sa2W, sa2b, sa2g, sa2be, Couts,
           fidx2, l2_xyz, bidx2, bufA, bufB, mean, var, l2_f); }

  // ---- feature enhance 1 ----
  run_fe(stream, l2_xyz, l2_f, fe1, xfe1, fe_x, fe_q, fe_k, fe_v, fe_sc, fe_xr, B, S2, 256, 64);

  // ---- bottleneck: max over S, add noise ----
  maxpool_kernel<<<g1(B * 256, 256), 256, 0, stream>>>(xfe1, xmax, B, S2, 256);
  add_noise_kernel<<<g1(B * 256, 256), 256, 0, stream>>>(xmax, noise, comp, out_comp, B * 256);

  // ---- decoder: (8,256) -> (8,32768) relu -> (1024,256) -> relu lin ----
  run_gemm(stream, comp, nullptr, decW1, decb1, nullptr, H, B, NP0 * 256, 256,
           256, 256, NP0 * 256, 1, 0, 0, 0, 0, 1, ACT_RELU);
  run_gemm(stream, H, nullptr, decW2, decb2, nullptr, dec_feat, B * NP0, 256, 256,
           256, 256, 256, 1, 0, 0, 0, 0, 1, ACT_RELU);

  // ---- coordinate head: 256 -> 128 -> 128 -> 3 ----
  run_gemm(stream, dec_feat, nullptr, coW1, cob1, nullptr, ct1, B * NP0, 128, 256,
           256, 256, 128, 1, 0, 0, 0, 0, 1, ACT_RELU);
  run_gemm(stream, ct1, nullptr, coW2, cob2, nullptr, ct2, B * NP0, 128, 128,
           128, 128, 128, 1, 0, 0, 0, 0, 1, ACT_RELU);
  run_gemm(stream, ct2, nullptr, coW3, cob3, nullptr, nxyz0, B * NP0, 3, 128,
           128, 128, 3, 1, 0, 0, 0, 0, 1, ACT_NONE);

  // ---- feature enhance 2 ----
  run_fe(stream, nxyz0, dec_feat, fe2, featU0, fe_x, fe_q, fe_k, fe_v, fe_sc, fe_xr, B, S2, 256, 64);

  // ---- 4x upsample: 128 -> 256 -> 512 -> 1024 -> 2048 points/batch ----
  const float* cxyz = nxyz0;
  const float* cfeat = featU0;
  int P = B * NP0;
  for (int s4 = 0; s4 < 4; ++s4) {
    run_gemm(stream, cfeat, nullptr, up[s4].Wo1, up[s4].bo1, nullptr, hbuf, P, 128, 256,
             256, 256, 128, 1, 0, 0, 0, 0, 1, ACT_RELU);
    run_gemm(stream, hbuf, nullptr, up[s4].Wo2, up[s4].bo2, nullptr, offbuf, P, 6, 128,
             128, 128, 6, 1, 0, 0, 0, 0, 1, ACT_TANH);
    float* child = (s4 & 1) ? childB : childA;
    run_gemm(stream, cfeat, nullptr, up[s4].Wf, up[s4].bf, nullptr, child, P, 512, 256,
             256, 256, 512, 1, 0, 0, 0, 0, 1, ACT_RELU);
    float* oxyz = (s4 == 3) ? out_xyz : ((s4 & 1) ? upxyzB : upxyzA);
    upsample_coords_kernel<<<g1(2L * P, 256), 256, 0, stream>>>(cxyz, offbuf, oxyz, P, 0.05f);
    cxyz = oxyz;
    cfeat = child;  // (P,512) reinterpreted flat as (2P,256)
    P *= 2;
  }

  // ---- chamfer distance between xyz (8,8192,3) and out_xyz (8,2048,3) ----
  zero_kernel<<<1, 32, 0, stream>>>(sums, 2);
  chamfer_min_kernel<<<dim3((NPTS + 63) / 64, B), 256, 0, stream>>>(xyz, out_xyz, NPTS, 2048, &sums[0]);
  chamfer_min_kernel<<<dim3((2048 + 63) / 64, B), 256, 0, stream>>>(out_xyz, xyz, 2048, NPTS, &sums[1]);
  chamfer_final_kernel<<<1, 32, 0, stream>>>(sums, out_cd,
                                             1.f / (float)(B * NPTS), 1.f / (float)(B * 2048));
}